// torch_solver_67559835566283
// MI455X (gfx1250) — compile-verified
//
#include <hip/hip_runtime.h>
#include <hip/hip_bf16.h>
#include <stdint.h>

// MI455X (gfx1250) implementation.
//
// Work: per-sample 256x256 slogdet * 1024 samples  (~11.5 GFLOP of LU work)
//       + elementwise non_infected*log(1-P) term   (bandwidth-trivial).
// Strategy: one workgroup (8 wave32) per sample; W lives entirely in LDS
// (256 x 257-float pitch = 257 KB, fits CDNA5's 320 KB/WGP LDS; the pitch
// makes column walks bank-conflict-free since 257 mod 64 == 1).
//  * P (sigmoid(A), diag=0) is staged to global once, then DMA'd into LDS by
//    the Tensor Data Mover: ONE tensor_load_to_lds per block, with TDM
//    pad_enable (interval=256 DW, amount=1 DW) materializing the 257-float
//    pitch in hardware. P lives in the 192 MB L2, so 1024 blocks pull it at
//    L2 bandwidth while their waves concurrently compute sf / r0.
//  * Blocked LU (NB=16, partial pivoting): rank-16 trailing updates run on
//    the matrix pipe with V_WMMA_F32_16X16X4_F32 (f32-precise), 4 chained
//    K=4 WMMAs per 16x16 tile; each wave owns a tile ROW so the negated L21
//    A-fragments are loaded once and reused across the row.
//  * Pivot argmax uses wave32 __shfl_xor cross-lane reduction (2 barriers/col).

#define NN   256
#define MM   1024
#define LDW  257       // 257 mod 64 == 1 -> conflict-free column access

typedef __attribute__((ext_vector_type(2))) float    v2f;
typedef __attribute__((ext_vector_type(8))) float    v8f;
typedef __attribute__((ext_vector_type(4))) uint32_t u32x4;
typedef __attribute__((ext_vector_type(8))) uint32_t u32x8;

// ---------------------------------------------------------------------------
// out[0] = 0
__global__ void zero_out_kernel(float* out) { out[0] = 0.0f; }

// ---------------------------------------------------------------------------
// P = sigmoid(A) with zero diagonal; accumulate -(sum non_infected*log(1-P+1e-10))
__global__ __launch_bounds__(256)
void prep_kernel(const float* __restrict__ A,
                 const float* __restrict__ noninf,
                 float* __restrict__ P,
                 float* __restrict__ out)
{
    const int idx = blockIdx.x * 256 + threadIdx.x;   // grid = 256 blocks
    const int i = idx >> 8;
    const int j = idx & 255;

    float a = A[idx];
    float p = 1.0f / (1.0f + __expf(-a));
    if (i == j) p = 0.0f;
    P[idx] = p;

    float term = noninf[idx] * __logf(1.0f - p + 1e-10f);

    // wave32 cross-lane sum, then 8-entry cross-wave sum
    #pragma unroll
    for (int off = 16; off > 0; off >>= 1)
        term += __shfl_xor(term, off, 32);

    __shared__ float red[8];
    if ((threadIdx.x & 31) == 0) red[threadIdx.x >> 5] = term;
    __syncthreads();
    if (threadIdx.x == 0) {
        float s = 0.0f;
        #pragma unroll
        for (int w = 0; w < 8; ++w) s += red[w];
        atomicAdd(out, -s);
    }
}

// ---------------------------------------------------------------------------
// One workgroup (256 threads = 8 wave32) per sample:
//   TDM-load P into LDS (padded pitch), build W, blocked LU w/ partial
//   pivoting (WMMA trailing updates), accumulate -log|det|.
__global__ __launch_bounds__(256)
void logdet_kernel(const float* __restrict__ P,
                   const int*   __restrict__ S,
                   float* __restrict__ out)
{
    extern __shared__ float W[];            // NN * LDW floats (dynamic LDS)
    __shared__ float sf[NN];
    __shared__ float redv[8];
    __shared__ int   redi[8];
    __shared__ int   pivs;
    __shared__ int   r0s;

    const int tid  = threadIdx.x;
    const int m    = blockIdx.x;
    const int lane = tid & 31;
    const int wave = tid >> 5;

    // ---- TDM: DMA P (256x256 f32) global -> LDS with 257-DWORD pitch ----
    // Issued by wave 0 only (TENSORcnt is per-wave; EXEC is ignored by TDM).
    if (wave == 0) {
        const uint64_t ga = (uint64_t)(uintptr_t)P;     // tile start == tensor start
        const uint32_t la = (uint32_t)(uintptr_t)W;     // LDS byte address of W

        u32x4 g0;
        g0[0] = 1u;                                     // count=1, no gather
        g0[1] = la;                                     // lds_addr
        g0[2] = (uint32_t)ga;                           // global_addr[31:0]
        g0[3] = (uint32_t)((ga >> 32) & 0x01FFFFFFu)    // global_addr[56:32]
              | (2u << 30);                             // type = 2 ("image")

        u32x8 g1;
        g1[0] = (2u << 16)        // data_size = 4 bytes
              | (1u << 20)        // pad_enable (load only)
              | (7u << 22);       // pad_interval = 256 DWORDs; pad_amount = 1 DWORD
        g1[1] = (uint32_t)NN << 16;   // tensor_dim0 = 256   (bits 79:48)
        g1[2] = (uint32_t)NN << 16;   // tensor_dim1 = 256   (bits 111:80)
        g1[3] = (uint32_t)NN << 16;   // tile_dim0   = 256   (bits 127:112)
        g1[4] = (uint32_t)NN;         // tile_dim1   = 256   (bits 143:128)
        g1[5] = (uint32_t)NN;         // tensor_dim0_stride = 256 (bits 207:160)
        g1[6] = 0u;
        g1[7] = 0u;

        asm volatile("tensor_load_to_lds %0, %1" :: "s"(g0), "s"(g1) : "memory");
    }

    // ---- meanwhile: mask row Sf, first infected index r0 (argmax) ----
    const int sv = S[m * NN + tid];
    sf[tid] = (float)sv;
    if (tid == 0) r0s = NN;
    __syncthreads();
    if (sv != 0) atomicMin(&r0s, tid);

    __builtin_amdgcn_s_wait_tensorcnt(0);   // wave 0 waits for TDM; others pass
    __syncthreads();
    const int r0 = (r0s < NN) ? r0s : 0;

    // ---- in-place: W = -sub = -(P * Sf_i * Sf_j) ----
    for (int idx = tid; idx < NN * NN; idx += 256) {
        const int i = idx >> 8;
        const int j = idx & 255;
        W[i * LDW + j] *= -sf[i] * sf[j];
    }
    __syncthreads();

    // ---- column sums -> diagonal; then row r0 := Sf ----
    float cs = 0.0f;
    for (int i = 0; i < NN; ++i) cs += W[i * LDW + tid];   // conflict-free
    __syncthreads();
    W[tid * LDW + tid] = (sf[tid] > 0.0f) ? -cs : 1.0f;    // diag of W = diagvals
    __syncthreads();
    W[r0 * LDW + tid] = sf[tid];
    __syncthreads();

    // ---- blocked LU, NB = 16 ----
    const int nsel = lane & 15;
    const int half = lane >> 4;     // 0 or 1
    float logacc = 0.0f;            // only tid 0 accumulates

    for (int kb = 0; kb < 16; ++kb) {
        const int p0 = kb * 16;

        // -------- panel factorization (16 columns, partial pivoting) --------
        for (int ci = 0; ci < 16; ++ci) {
            const int cc = p0 + ci;

            // argmax |W[i][cc]| over i in [cc, NN): wave-level shfl reduction
            float v   = (tid >= cc) ? fabsf(W[tid * LDW + cc]) : -1.0f;
            int   idx = tid;
            #pragma unroll
            for (int off = 16; off > 0; off >>= 1) {
                const float ov = __shfl_xor(v,   off, 32);
                const int   oi = __shfl_xor(idx, off, 32);
                if (ov > v) { v = ov; idx = oi; }
            }
            if (lane == 0) { redv[wave] = v; redi[wave] = idx; }
            __syncthreads();
            if (tid == 0) {
                float bv = redv[0]; int bi = redi[0];
                #pragma unroll
                for (int w = 1; w < 8; ++w)
                    if (redv[w] > bv) { bv = redv[w]; bi = redi[w]; }
                pivs = bi;
            }
            __syncthreads();
            const int piv = pivs;

            // swap full rows cc <-> piv (|det| unchanged); thread owns col tid
            if (piv != cc) {
                const float t1 = W[cc  * LDW + tid];
                const float t2 = W[piv * LDW + tid];
                W[cc  * LDW + tid] = t2;
                W[piv * LDW + tid] = t1;
            }
            __syncthreads();

            const float d = W[cc * LDW + cc];
            if (tid == 0) logacc += __logf(fabsf(d) + 1e-30f);
            const float inv = 1.0f / d;
            if (tid > cc) W[tid * LDW + cc] *= inv;        // L column
            __syncthreads();

            // rank-1 update inside remaining panel columns
            if (tid > cc) {
                const float lik = W[tid * LDW + cc];
                for (int j = cc + 1; j < p0 + 16; ++j)
                    W[tid * LDW + j] -= lik * W[cc * LDW + j];
            }
            __syncthreads();
        }

        const int t0 = p0 + 16;
        const int T  = NN - t0;
        if (T > 0) {
            // -------- U12 = L11^{-1} * A12 (unit lower triangular solve) ----
            if (tid < T) {
                const int j = t0 + tid;
                #pragma unroll
                for (int r = 1; r < 16; ++r) {
                    float s = 0.0f;
                    for (int q = 0; q < r; ++q)
                        s += W[(p0 + r) * LDW + p0 + q] * W[(p0 + q) * LDW + j];
                    W[(p0 + r) * LDW + j] -= s;
                }
            }
            __syncthreads();

            // -------- trailing update  A22 -= L21 * U12  via f32 WMMA --------
            // Each wave owns full tile ROWS: the (negated) L21 A-fragments are
            // loaded once per row and reused across all nT column tiles.
            // Tile-row loop is wave-uniform so EXEC stays all-ones (WMMA req).
            const int nT = T >> 4;
            for (int ti = wave; ti < nT; ti += 8) {
                const int I = t0 + ti * 16;

                // A fragments: 16x16 L21 block, negated (f32 WMMA lacks A-neg)
                v2f a0, a1, a2, a3;
                {
                    const float* arow = &W[(I + nsel) * LDW + p0 + half * 2];
                    a0.x = -arow[0];  a0.y = -arow[1];
                    a1.x = -arow[4];  a1.y = -arow[5];
                    a2.x = -arow[8];  a2.y = -arow[9];
                    a3.x = -arow[12]; a3.y = -arow[13];
                }

                for (int tj = 0; tj < nT; ++tj) {
                    const int J = t0 + tj * 16;

                    // C tile: lane n = nsel, VGPR r -> row I + half*8 + r
                    v8f c;
                    #pragma unroll
                    for (int r = 0; r < 8; ++r)
                        c[r] = W[(I + half * 8 + r) * LDW + J + nsel];

                    // 4 chained K=4 WMMAs cover the K=16 panel
                    #pragma unroll
                    for (int kk = 0; kk < 4; ++kk) {
                        const int bk = kk * 4 + half * 2;
                        v2f b;
                        b.x = W[(p0 + bk)     * LDW + J + nsel];
                        b.y = W[(p0 + bk + 1) * LDW + J + nsel];
                        const v2f a = (kk == 0) ? a0 : (kk == 1) ? a1
                                    : (kk == 2) ? a2 : a3;
                        c = __builtin_amdgcn_wmma_f32_16x16x4_f32(
                                false, a, false, b, (short)0, c, false, false);
                    }

                    #pragma unroll
                    for (int r = 0; r < 8; ++r)
                        W[(I + half * 8 + r) * LDW + J + nsel] = c[r];
                }
            }
            __syncthreads();
        }
    }

    if (tid == 0) atomicAdd(out, -logacc);
}

// ---------------------------------------------------------------------------
extern "C" void kernel_launch(void* const* d_in, const int* in_sizes, int n_in,
                              void* d_out, int out_size, void* d_ws, size_t ws_size,
                              hipStream_t stream) {
    (void)in_sizes; (void)n_in; (void)out_size; (void)ws_size;
    const float* A      = (const float*)d_in[0];   // [256,256] f32
    const float* noninf = (const float*)d_in[1];   // [256,256] f32
    const int*   S      = (const int*)  d_in[2];   // [1024,256] i32
    float* out = (float*)d_out;                    // scalar f32
    float* P   = (float*)d_ws;                     // 256 KB scratch for P

    zero_out_kernel<<<1, 1, 0, stream>>>(out);
    prep_kernel<<<NN, 256, 0, stream>>>(A, noninf, P, out);

    const size_t ldsBytes = (size_t)NN * LDW * sizeof(float);  // ~257 KB < 320 KB/WGP
    logdet_kernel<<<MM, 256, ldsBytes, stream>>>(P, S, out);
}